// DEAM_5076651343977
// MI455X (gfx1250) — compile-verified
//
#include <hip/hip_runtime.h>
#include <math.h>

// ---------------------------------------------------------------------------
// DEAM attention, MI455X (gfx1250), wave32, fp32 WMMA 16x16x4.
// B=8, C=64, H=W=256, DS=8 -> h=w=32, N=1024.
//
// Roofline: ~2.4 GFLOP total vs >=400 MB streaming HBM traffic -> memory
// bound on x/diff/out. Attention working set (~12 MB) is kept L2-resident;
// all full-resolution streams use non-temporal hints so they don't evict it.
// ---------------------------------------------------------------------------

typedef float v2f __attribute__((ext_vector_type(2)));
typedef float v4f __attribute__((ext_vector_type(4)));
typedef float v8f __attribute__((ext_vector_type(8)));

#define WMMA_F32(a, b, c) \
  __builtin_amdgcn_wmma_f32_16x16x4_f32(false, (a), false, (b), (short)0, (c), false, false)

constexpr int BB = 8, CC = 64, HH = 256, WWD = 256, NN = 1024; // NN = 32*32 pooled tokens

// ---------------------------------------------------------------------------
// 8x8 average pool:  in [B,C,256,256] -> out [B,C,1024]  (n = yo*32+xo)
// Streaming reads are non-temporal (134 MB each, zero reuse); the 2 MB pooled
// output is regular-temporal so it stays hot in L2 for the GEMM kernels.
// ---------------------------------------------------------------------------
__global__ void pool8_kernel(const float* __restrict__ in, float* __restrict__ out) {
  int idx = blockIdx.x * blockDim.x + threadIdx.x;   // exactly B*C*N threads
  int n  = idx & (NN - 1);
  int bc = idx >> 10;
  int yo = n >> 5, xo = n & 31;
  const v4f* p = (const v4f*)(in + ((size_t)bc * HH + (size_t)yo * 8) * WWD +
                              (size_t)xo * 8);
  float s = 0.f;
#pragma unroll
  for (int r = 0; r < 8; ++r) {
    v4f u = __builtin_nontemporal_load(p + (size_t)r * (WWD / 4));
    v4f v = __builtin_nontemporal_load(p + (size_t)r * (WWD / 4) + 1);
    s += u.x + u.y + u.z + u.w + v.x + v.y + v.z + v.w;
  }
  out[idx] = s * (1.0f / 64.0f);
}

// ---------------------------------------------------------------------------
// 1x1 conv as GEMM:  dst = Wm[64x64] @ src[64x1024] + bias   (per batch)
// TRANSPOSE_OUT==0: dst[b][c][n]   (row-major C x N)
// TRANSPOSE_OUT==1: dst[b][n][c]   (vT layout for the output GEMM)
// grid: (N/128, B), block 256 (8 waves). Wave w: mtile w&3, n-half w>>2.
// ---------------------------------------------------------------------------
template <int TRANSPOSE_OUT>
__global__ void qkv_kernel(const float* __restrict__ Wm, const float* __restrict__ bias,
                           const float* __restrict__ src, float* __restrict__ dst) {
  const int b    = blockIdx.y;
  const int lane = threadIdx.x & 31;
  const int wave = threadIdx.x >> 5;
  const int l16  = lane & 15;
  const int kr   = (lane < 16) ? 0 : 2;   // ISA A/B frag: lanes 16-31 hold K+2 / K+3
  const int hi8  = (lane < 16) ? 0 : 8;   // ISA C/D frag: lanes 16-31 hold M+8
  const int m0   = (wave & 3) * 16;
  const int nbase = blockIdx.x * 128 + (wave >> 2) * 64;
  const float* srcb = src + (size_t)b * CC * NN;

  // hoist all 16 A-fragments of the 64x64 weight (reused across 4 n-tiles)
  v2f Af[16];
#pragma unroll
  for (int ks = 0; ks < 16; ++ks) {
    int k0 = ks * 4 + kr;
    int row = m0 + l16;
    Af[ks].x = Wm[row * 64 + k0];
    Af[ks].y = Wm[row * 64 + k0 + 1];
  }

  for (int nt = 0; nt < 4; ++nt) {
    int n0 = nbase + nt * 16;
    v8f acc = {};
#pragma unroll
    for (int ks = 0; ks < 16; ++ks) {
      int k0 = ks * 4 + kr;
      v2f Bf;
      Bf.x = srcb[(size_t)k0 * NN + n0 + l16];
      Bf.y = srcb[(size_t)(k0 + 1) * NN + n0 + l16];
      acc = WMMA_F32(Af[ks], Bf, acc);
    }
#pragma unroll
    for (int r = 0; r < 8; ++r) {
      int row = m0 + r + hi8;
      float val = acc[r] + bias[row];
      int col = n0 + l16;
      if (TRANSPOSE_OUT)
        dst[(size_t)b * NN * CC + (size_t)col * CC + row] = val;
      else
        dst[(size_t)b * CC * NN + (size_t)row * NN + col] = val;
    }
  }
}

// ---------------------------------------------------------------------------
// Fused attention per (batch, 32-row n-tile):
//   E = (q^T k) * C^-0.5 -> softmax rows -> out_t = attn @ vT
// grid: (N/32, B), block 256 (8 waves).  LDS: 32 x 1025 f32 (odd pitch ->
// conflict-free column reads of A-fragments in phase 3).
// ---------------------------------------------------------------------------
__global__ void attn_kernel(const float* __restrict__ q, const float* __restrict__ k,
                            const float* __restrict__ vT, float* __restrict__ outs) {
  __shared__ float E[32][1025];
  const int b    = blockIdx.y;
  const int nbase = blockIdx.x * 32;
  const int lane = threadIdx.x & 31;
  const int wave = threadIdx.x >> 5;
  const int l16  = lane & 15;
  const int kr   = (lane < 16) ? 0 : 2;
  const int hi8  = (lane < 16) ? 0 : 8;
  const float* qb = q  + (size_t)b * CC * NN;
  const float* kb = k  + (size_t)b * CC * NN;
  const float* vb = vT + (size_t)b * NN * CC;

  // ---- phase 1: energy tile [32n x 1024m], K = C = 64 -----------------------
  {
    const int nloc  = (wave & 1) * 16;       // local n sub-tile
    const int mhalf = (wave >> 1) * 256;     // this wave's 256-wide m range
    v2f Af[16];
#pragma unroll
    for (int ks = 0; ks < 16; ++ks) {
      int c0 = ks * 4 + kr;
      Af[ks].x = qb[(size_t)c0 * NN + nbase + nloc + l16];
      Af[ks].y = qb[(size_t)(c0 + 1) * NN + nbase + nloc + l16];
    }
    for (int mt = 0; mt < 16; ++mt) {
      int m0 = mhalf + mt * 16;
      v8f acc = {};
#pragma unroll
      for (int ks = 0; ks < 16; ++ks) {
        int c0 = ks * 4 + kr;
        v2f Bf;
        Bf.x = kb[(size_t)c0 * NN + m0 + l16];
        Bf.y = kb[(size_t)(c0 + 1) * NN + m0 + l16];
        acc = WMMA_F32(Af[ks], Bf, acc);
      }
#pragma unroll
      for (int r = 0; r < 8; ++r)
        E[nloc + r + hi8][m0 + l16] = acc[r] * 0.125f;   // * C^-0.5
    }
  }
  __syncthreads();

  // ---- phase 2: stable softmax over each of the 32 rows (1024 elems) --------
  {
#pragma unroll
    for (int rr = 0; rr < 4; ++rr) {
      int row = wave * 4 + rr;
      float mx = -3.4e38f;
#pragma unroll
      for (int j = 0; j < 32; ++j) mx = fmaxf(mx, E[row][lane + 32 * j]);
#pragma unroll
      for (int off = 16; off > 0; off >>= 1) mx = fmaxf(mx, __shfl_xor(mx, off, 32));
      float s = 0.f;
#pragma unroll
      for (int j = 0; j < 32; ++j) {
        float e = __expf(E[row][lane + 32 * j] - mx);
        E[row][lane + 32 * j] = e;
        s += e;
      }
#pragma unroll
      for (int off = 16; off > 0; off >>= 1) s += __shfl_xor(s, off, 32);
      float inv = 1.0f / s;
#pragma unroll
      for (int j = 0; j < 32; ++j) E[row][lane + 32 * j] *= inv;
    }
  }
  __syncthreads();

  // ---- phase 3: out_t[32n x 64c] = attn[32x1024] @ vT[1024x64] --------------
  {
    const int nloc = (wave & 1) * 16;
    const int c0   = (wave >> 1) * 16;
    v8f acc = {};
#pragma unroll 4
    for (int ks = 0; ks < 256; ++ks) {
      int k0 = ks * 4 + kr;
      v2f Aa, Bf;
      Aa.x = E[nloc + l16][k0];
      Aa.y = E[nloc + l16][k0 + 1];
      Bf.x = vb[(size_t)k0 * CC + c0 + l16];
      Bf.y = vb[(size_t)(k0 + 1) * CC + c0 + l16];
      acc = WMMA_F32(Aa, Bf, acc);
    }
#pragma unroll
    for (int r = 0; r < 8; ++r) {
      int ng = nbase + nloc + r + hi8;
      outs[(size_t)b * NN * CC + (size_t)ng * CC + c0 + l16] = acc[r];
    }
  }
}

// ---------------------------------------------------------------------------
// Nearest upsample x8 + residual add:  out = x + repeat(out_small)
// out_small layout [B][N][C].  One v4f of x per thread (4 consecutive x
// pixels always share the same pooled cell since 4 | 8).
// x read + out write are non-temporal streams; out_small reads stay RT
// (2 MB, each element reused 64x, L2-resident).
// ---------------------------------------------------------------------------
__global__ void upsample_add_kernel(const float* __restrict__ xin,
                                    const float* __restrict__ outs,
                                    float* __restrict__ out) {
  int idx = blockIdx.x * blockDim.x + threadIdx.x;  // one v4f
  int e0 = idx * 4;
  int xq = e0 & 255;
  int y  = (e0 >> 8) & 255;
  int bc = e0 >> 16;
  int c = bc & 63;
  int b = bc >> 6;
  int n = ((y >> 3) << 5) + (xq >> 3);
  float add = outs[((size_t)b * NN + n) * CC + c];
  v4f v = __builtin_nontemporal_load((const v4f*)xin + idx);
  v.x += add; v.y += add; v.z += add; v.w += add;
  __builtin_nontemporal_store(v, (v4f*)out + idx);
}

// ---------------------------------------------------------------------------
extern "C" void kernel_launch(void* const* d_in, const int* in_sizes, int n_in,
                              void* d_out, int out_size, void* d_ws, size_t ws_size,
                              hipStream_t stream) {
  const float* x    = (const float*)d_in[0];
  const float* diff = (const float*)d_in[1];
  const float* Wq   = (const float*)d_in[2];
  const float* bq   = (const float*)d_in[3];
  const float* Wk   = (const float*)d_in[4];
  const float* bk   = (const float*)d_in[5];
  const float* Wv   = (const float*)d_in[6];
  const float* bv   = (const float*)d_in[7];
  float* out = (float*)d_out;

  char* ws = (char*)d_ws;
  const size_t SZ = (size_t)BB * CC * NN * sizeof(float);  // 2 MB per buffer
  float* dpool  = (float*)(ws + 0 * SZ);   // pooled diff   [B][C][N]
  float* pxpool = (float*)(ws + 1 * SZ);   // pooled x      [B][C][N]
  float* qbuf   = (float*)(ws + 2 * SZ);   // q             [B][C][N]
  float* kbuf   = (float*)(ws + 3 * SZ);   // k             [B][C][N]
  float* vTbuf  = (float*)(ws + 4 * SZ);   // v transposed  [B][N][C]
  float* osmall = (float*)(ws + 5 * SZ);   // attn output   [B][N][C]

  dim3 blk(256);
  pool8_kernel<<<BB * CC * NN / 256, blk, 0, stream>>>(diff, dpool);
  pool8_kernel<<<BB * CC * NN / 256, blk, 0, stream>>>(x, pxpool);

  dim3 gq(NN / 128, BB);
  qkv_kernel<0><<<gq, blk, 0, stream>>>(Wq, bq, dpool,  qbuf);
  qkv_kernel<0><<<gq, blk, 0, stream>>>(Wk, bk, dpool,  kbuf);
  qkv_kernel<1><<<gq, blk, 0, stream>>>(Wv, bv, pxpool, vTbuf);

  dim3 ga(NN / 32, BB);
  attn_kernel<<<ga, blk, 0, stream>>>(qbuf, kbuf, vTbuf, osmall);

  upsample_add_kernel<<<(BB * CC * HH * WWD / 4) / 256, blk, 0, stream>>>(x, osmall, out);
}